// magrhoxx_TwoBandModel_53163105189913
// MI455X (gfx1250) — compile-verified
//
#include <hip/hip_runtime.h>

// ---------------------------------------------------------------------------
// Two-band magnetoresistance rho_xx — streaming elementwise map.
//
// Single-reciprocal form:  rho_xx = A*Q * rcp(A^2 + C^2)
//   d1 = 1 + mu1^2 B^2,  d2 = 1 + mu2^2 B^2,  Q = d1*d2
//   A  = s1*d2 + s2*d1             (= sxx * Q)
//   C  = (s1*mu1*d2 + s2*mu2*d1)*B (= (s1xy+s2xy) * Q)
//
// Divide implemented as v_rcp_f32 + 1 Newton step + mul (~1 ulp; denominator
// is bounded in [~1e-5, ~1e1] for this model, so no range scaling needed).
// Bandwidth-bound (8 B/elem): b128 loads/stores, RT cache hints (128 MiB
// working set is L2-resident in the 192 MB L2 across graph replays), and
// gfx1250 global_prefetch_b8 for HBM latency hiding.
// WMMA deliberately unused: no matrix-shaped work exists in this op.
// ---------------------------------------------------------------------------

#define ELECTRON_F 1.602176634e-19f

__device__ __forceinline__ float fast_div(float num, float den)
{
    float r = __builtin_amdgcn_rcpf(den);      // v_rcp_f32 (<= 1 ulp)
    const float e = fmaf(-den, r, 1.0f);       // Newton-Raphson residual
    r = fmaf(r, e, r);                         // refined reciprocal (~0.5 ulp)
    return num * r;
}

__device__ __forceinline__ float rho_xx_elem(float b,
                                             float sigma1, float m1sq, float s1m1,
                                             float sigma2, float m2sq, float s2m2)
{
    const float bsq = b * b;
    const float d1  = fmaf(m1sq, bsq, 1.0f);
    const float d2  = fmaf(m2sq, bsq, 1.0f);
    const float Q   = d1 * d2;
    const float A   = fmaf(sigma1, d2, sigma2 * d1);     // sxx * Q
    const float C   = fmaf(s1m1,  d2, s2m2  * d1) * b;   // (s1xy+s2xy) * Q
    const float num = A * Q;
    const float den = fmaf(A, A, C * C);                 // (sxx^2 + sxy^2) * Q^2
    return fast_div(num, den);
}

__global__ __launch_bounds__(256) void magrhoxx_TwoBandModel_53163105189913_kernel(
    const float* __restrict__ params,
    const float* __restrict__ B,
    float* __restrict__ out,
    int N)
{
    // Uniform parameter loads -> scalar FP pipeline (s_mul_f32 etc.).
    const float n1     = params[0] * 1e16f;
    const float mu1    = params[1];
    const float n2     = params[2] * 1e16f;
    const float mu2    = params[3];
    // params[4] (sigmaA) enters only as sigmaA*0.0 == +/-0 -> exact no-op.

    const float sigma1 = ELECTRON_F * n1 * fabsf(mu1);
    const float sigma2 = ELECTRON_F * n2 * fabsf(mu2);
    const float m1sq   = mu1 * mu1;
    const float m2sq   = mu2 * mu2;
    const float s1m1   = sigma1 * mu1;
    const float s2m2   = sigma2 * mu2;

    const int tid      = blockIdx.x * blockDim.x + threadIdx.x;
    const int nthreads = gridDim.x * blockDim.x;
    const int n4       = N >> 2;          // number of float4 packets
    const int stride2  = nthreads * 2;    // two packets per iteration

    const float4* __restrict__ B4 = reinterpret_cast<const float4*>(B);
    float4*       __restrict__ O4 = reinterpret_cast<float4*>(out);

    // Two coalesced b128 loads in flight per iteration; prefetch the next
    // iteration's packets (gfx1250 global_prefetch_b8, RT -> park in L2).
    for (int i = tid; i < n4; i += stride2) {
        const int j = i + nthreads;

        const int pf0 = i + stride2;
        const int pf1 = j + stride2;
        if (pf0 < n4) __builtin_prefetch(&B4[pf0], 0, 3);
        if (pf1 < n4) __builtin_prefetch(&B4[pf1], 0, 3);

        const float4 b0 = B4[i];
        const bool   hasj = (j < n4);
        float4 b1;
        if (hasj) b1 = B4[j];

        float4 r0;
        r0.x = rho_xx_elem(b0.x, sigma1, m1sq, s1m1, sigma2, m2sq, s2m2);
        r0.y = rho_xx_elem(b0.y, sigma1, m1sq, s1m1, sigma2, m2sq, s2m2);
        r0.z = rho_xx_elem(b0.z, sigma1, m1sq, s1m1, sigma2, m2sq, s2m2);
        r0.w = rho_xx_elem(b0.w, sigma1, m1sq, s1m1, sigma2, m2sq, s2m2);
        O4[i] = r0;

        if (hasj) {
            float4 r1;
            r1.x = rho_xx_elem(b1.x, sigma1, m1sq, s1m1, sigma2, m2sq, s2m2);
            r1.y = rho_xx_elem(b1.y, sigma1, m1sq, s1m1, sigma2, m2sq, s2m2);
            r1.z = rho_xx_elem(b1.z, sigma1, m1sq, s1m1, sigma2, m2sq, s2m2);
            r1.w = rho_xx_elem(b1.w, sigma1, m1sq, s1m1, sigma2, m2sq, s2m2);
            O4[j] = r1;
        }
    }

    // Scalar tail for N not divisible by 4.
    const int tail_start = n4 << 2;
    for (int i = tail_start + tid; i < N; i += nthreads) {
        out[i] = rho_xx_elem(B[i], sigma1, m1sq, s1m1, sigma2, m2sq, s2m2);
    }
}

extern "C" void kernel_launch(void* const* d_in, const int* in_sizes, int n_in,
                              void* d_out, int out_size, void* d_ws, size_t ws_size,
                              hipStream_t stream)
{
    (void)n_in; (void)d_ws; (void)ws_size; (void)out_size;

    const float* params = reinterpret_cast<const float*>(d_in[0]); // [5]
    const float* B      = reinterpret_cast<const float*>(d_in[1]); // [N]
    float*       out    = reinterpret_cast<float*>(d_out);         // [N]

    const int N  = in_sizes[1];
    const int n4 = N >> 2;

    const int threads    = 256;   // 8 wave32s per block
    const int per_thread = 8;     // 8 b128 packets per thread (4 iterations x 2)
    long long want = ((long long)n4 + (long long)threads * per_thread - 1) /
                     ((long long)threads * per_thread);
    if (want < 1) want = 1;
    if (want > 131072) want = 131072;
    const int blocks = (int)want; // N=16.7M -> 2048 blocks

    magrhoxx_TwoBandModel_53163105189913_kernel<<<blocks, threads, 0, stream>>>(
        params, B, out, N);
}